// QuantumGateLayer_65481071404100
// MI455X (gfx1250) — compile-verified
//
#include <hip/hip_runtime.h>

typedef float v8f __attribute__((ext_vector_type(8)));
typedef float v2f __attribute__((ext_vector_type(2)));

// out[b] = prod_{w=0..15} cos(x[b,w] + p[w])   (closed form of the circuit)
// One wave32 per 16 samples. Per-sample product evaluated as
//   sign * exp2( sum_w log2|cos t_w| )
// with the 16-wide sum done by 4x V_WMMA_F32_16X16X4_F32 (B = ones).
// Hardware transcendentals (v_cos_f32 / v_log_f32 / v_exp_f32) only:
// |t| = |x+p| is small (x,p ~ N(0,1)), so no libm range-reduction needed.
__global__ __launch_bounds__(32) void qgate_closed_form_wmma(
    const float* __restrict__ x,      // (BSZ, 16)
    const float* __restrict__ p,      // (16,)
    float* __restrict__ out)          // (BSZ, 1)
{
    const int lane   = threadIdx.x;          // 0..31 (wave32)
    const int m      = lane & 15;            // row (sample within 16x16 tile)
    const int kBase  = (lane >> 4) << 1;     // lanes 0-15 -> K=0,1 ; lanes 16-31 -> K=2,3
    const int sample = blockIdx.x * 16 + m;

    v8f acc = {};                            // 16x16 f32 accumulator (row sums)
    const v2f bones = {1.0f, 1.0f};          // B = all-ones 4x16
    float sgn = 1.0f;

    // 4 K-chunks of 4 wires each -> covers wires 0..15.
    // This lane's two wires per chunk are adjacent & even-aligned -> float2 load.
    #pragma unroll
    for (int c = 0; c < 4; ++c) {
        const int w0 = 4 * c + kBase;
        const float2 xv = *reinterpret_cast<const float2*>(x + sample * 16 + w0);
        const float2 pv = *reinterpret_cast<const float2*>(p + w0);
        const float c0 = __cosf(xv.x + pv.x);    // v_cos_f32
        const float c1 = __cosf(xv.y + pv.y);
        sgn *= (c0 < 0.0f ? -1.0f : 1.0f);
        sgn *= (c1 < 0.0f ? -1.0f : 1.0f);
        v2f a;
        a.x = __log2f(__builtin_fabsf(c0));      // v_log_f32; -inf if cos==0 -> exp2 -> 0
        a.y = __log2f(__builtin_fabsf(c1));
        // D = A x B + C : every column of D gets the row-sum of the A-chunk
        acc = __builtin_amdgcn_wmma_f32_16x16x4_f32(
            /*neg_a=*/false, a, /*neg_b=*/false, bones,
            /*c_mod=*/(short)0, acc, /*reuse_a=*/false, /*reuse_b=*/false);
    }

    // Combine sign parity across the K-split lane pair (lane <-> lane^16):
    // afterwards lane i (and lane i+16) both hold the full 16-wire sign of row i.
    sgn *= __shfl_xor(sgn, 16, 32);
    const unsigned negmask = (unsigned)__ballot(sgn < 0.0f);

    // D layout: VGPR j -> lanes 0-15: (M=j, N=lane) ; lanes 16-31: (M=8+j, N=lane-16)
    // Lane 0 holds D[0..7][0], lane 16 holds D[8..15][0].
    if ((lane & 15) == 0) {
        const int rowBase = (lane >> 4) << 3;            // 0 or 8
        const int outBase = blockIdx.x * 16 + rowBase;
        #pragma unroll
        for (int j = 0; j < 8; ++j) {
            const int row = rowBase + j;
            const float s = ((negmask >> row) & 1u) ? -1.0f : 1.0f;
            out[outBase + j] = s * __builtin_amdgcn_exp2f(acc[j]);   // v_exp_f32
        }
    }
}

extern "C" void kernel_launch(void* const* d_in, const int* in_sizes, int n_in,
                              void* d_out, int out_size, void* d_ws, size_t ws_size,
                              hipStream_t stream) {
    (void)in_sizes; (void)n_in; (void)d_ws; (void)ws_size;
    const float* x = (const float*)d_in[0];   // (256, 16) f32
    const float* p = (const float*)d_in[1];   // (16,)     f32
    float* out = (float*)d_out;               // (256, 1)  f32

    const int samples = out_size;             // 256
    const int blocks  = samples / 16;         // one wave per 16 samples
    qgate_closed_form_wmma<<<dim3(blocks), dim3(32), 0, stream>>>(x, p, out);
}